// Solver_61632780698035
// MI455X (gfx1250) — compile-verified
//
#include <hip/hip_runtime.h>

typedef float v4f __attribute__((ext_vector_type(4)));
typedef int   v4i __attribute__((ext_vector_type(4)));

#if defined(__has_builtin)
#  if __has_builtin(__builtin_amdgcn_global_load_async_to_lds_b128)
#    define USE_ASYNC_LDS 1
#  endif
#endif
#ifndef USE_ASYNC_LDS
#  define USE_ASYNC_LDS 0
#endif

namespace {
constexpr int kH  = 480;
constexpr int kW  = 640;
constexpr int kHW = kH * kW;
constexpr float kEps = 1e-18f;
constexpr int kPlanes = 24;              // 3 rows x 8 channels
}

#if USE_ASYNC_LDS
typedef __attribute__((address_space(1))) v4i* as1_v4ip;   // global
typedef __attribute__((address_space(3))) v4i* as3_v4ip;   // LDS
#endif

__global__ __launch_bounds__(160) void prop_kernel(
    const float* __restrict__ depth, const float* __restrict__ normal,
    const float* __restrict__ rgb,   const float* __restrict__ conf,
    const float* __restrict__ kinv,  float* __restrict__ out)
{
    const int y  = blockIdx.x;          // row
    const int b  = blockIdx.y;          // batch
    const int oi = blockIdx.z;          // 0,1,2 -> o = 1,3,5
    const int o  = 2 * oi + 1;
    const int x0 = threadIdx.x * 4;     // 4 consecutive pixels per thread

    const float* D = depth  + (size_t)b * kHW;
    const float* N = normal + (size_t)b * 3 * kHW;
    const float* R = rgb    + (size_t)b * 3 * kHW;
    const float* C = conf   + (size_t)b * kHW;
    const float* K = kinv   + (size_t)b * 9;

    // LDS staging: [row(3)][channel(8)][kW] floats = 60 KB.
    // channels: 0=depth 1..3=normal 4=conf 5..7=rgb, rows: 0=y-o 1=y 2=y+o
    __shared__ float lds[kPlanes * kW];

    {
        const float* pb[8] = {D, N, N + kHW, N + 2 * kHW, C, R, R + kHW, R + 2 * kHW};
#pragma unroll
        for (int r = 0; r < 3; ++r) {
            int yr = y + (r - 1) * o;
            yr = (yr < 0) ? 0 : (yr >= kH ? kH - 1 : yr);   // OOB rows staged clamped, never used
#pragma unroll
            for (int ch = 0; ch < 8; ++ch) {
                const float* g = pb[ch] + (size_t)yr * kW + x0;
                float*       l = &lds[(r * 8 + ch) * kW + x0];
#if USE_ASYNC_LDS
                __builtin_amdgcn_global_load_async_to_lds_b128(
                    (as1_v4ip)(v4i*)(const_cast<float*>(g)),   // global src (AS1)
                    (as3_v4ip)(v4i*)(l),                       // LDS dst (AS3 via addrspacecast)
                    0, 0);
#else
                *(v4f*)l = *(const v4f*)g;                  // global_load_b128 + ds_store_b128
#endif
            }
        }
#if USE_ASYNC_LDS
#  if __has_builtin(__builtin_amdgcn_s_wait_asynccnt)
        __builtin_amdgcn_s_wait_asynccnt(0);
#  else
        asm volatile("s_wait_asynccnt 0x0" ::: "memory");
#  endif
#endif
        __syncthreads();
    }

    const float k0=K[0],k1=K[1],k2=K[2],k3=K[3],k4=K[4],k5=K[5],k6=K[6],k7=K[7],k8=K[8];

    // Back-projected rays at the 4 output pixels: ray(x,y) = K_inv @ (x, y, 1)
    float rx[4], ry[4], rz[4];
#pragma unroll
    for (int j = 0; j < 4; ++j) {
        const float xf = (float)(x0 + j), yf = (float)y;
        rx[j] = fmaf(k0, xf, fmaf(k1, yf, k2));
        ry[j] = fmaf(k3, xf, fmaf(k4, yf, k5));
        rz[j] = fmaf(k6, xf, fmaf(k7, yf, k8));
    }

    // Center depth (boundary fallback): aligned vector read from staged row 1.
    const v4f dc = *(const v4f*)&lds[(1 * 8 + 0) * kW + x0];

    // Clamped x indices for the shifted columns (values selected away when OOB).
    int xm[4], xp[4];
#pragma unroll
    for (int j = 0; j < 4; ++j) {
        int a = x0 + j - o;  xm[j] = (a < 0) ? 0 : a;                 // x-o >= 0 clamp
        int c = x0 + j + o;  xp[j] = (c >= kW) ? (kW - 1) : c;        // x+o < kW clamp
    }

    // Combo order matches reference: per o: [ud d1,d2, lr d1,d2, lu d1,d2, ld d1,d2]
    const int dysT[8] = {-1, +1,  0,  0, -1, +1, -1, +1};
    const int dxsT[8] = { 0,  0, -1, +1, -1, +1, +1, -1};

#pragma unroll
    for (int dir = 0; dir < 8; ++dir) {
        const int sdy = dysT[dir], sdx = dxsT[dir];
        const int dy = sdy * o, dx = sdx * o;
        const int ys = y + dy;
        const bool yin = ((unsigned)ys < (unsigned)kH);               // block-uniform
        const int r = 1 + sdy;                                        // LDS row index
        // ray(xs,ys) = ray(x,y) + dx*col0(Kinv) + dy*col1(Kinv)
        const float ox = dx * k0 + dy * k1;
        const float oy = dx * k3 + dy * k4;
        const float oz = dx * k6 + dy * k7;

        v4f dv, cv, r0v, r1v, r2v;

        if (sdx == 0) {
            // dx == 0: sources are the aligned 4-px segment -> ds_load_b128 per channel.
            const int pbase = (r * 8) * kW + x0;
            const v4f sd  = *(const v4f*)&lds[pbase];
            const v4f sn0 = *(const v4f*)&lds[pbase + 1 * kW];
            const v4f sn1 = *(const v4f*)&lds[pbase + 2 * kW];
            const v4f sn2 = *(const v4f*)&lds[pbase + 3 * kW];
            const v4f sc  = *(const v4f*)&lds[pbase + 4 * kW];
            const v4f sg0 = *(const v4f*)&lds[pbase + 5 * kW];
            const v4f sg1 = *(const v4f*)&lds[pbase + 6 * kW];
            const v4f sg2 = *(const v4f*)&lds[pbase + 7 * kW];
#pragma unroll
            for (int j = 0; j < 4; ++j) {
                const float n0 = sn0[j], n1 = sn1[j], n2 = sn2[j];
                const float den = fmaf(n0, rx[j], fmaf(n1, ry[j], fmaf(n2, rz[j], kEps)));
                const float nom = sd[j] * fmaf(n0, rx[j] + ox,
                                          fmaf(n1, ry[j] + oy, n2 * (rz[j] + oz)));
                float v = nom * __builtin_amdgcn_rcpf(den);
                v = fminf(fmaxf(v, 0.01f), 10.0f);
                dv[j]  = yin ? v      : dc[j];
                cv[j]  = yin ? sc[j]  : 0.0f;
                r0v[j] = yin ? sg0[j] : 0.0f;
                r1v[j] = yin ? sg1[j] : 0.0f;
                r2v[j] = yin ? sg2[j] : 0.0f;
            }
        } else {
#pragma unroll
            for (int j = 0; j < 4; ++j) {
                const int x  = x0 + j;
                const int xs = x + dx;
                const int xsel = (sdx < 0) ? xm[j] : xp[j];
                const bool xin = ((unsigned)xs < (unsigned)kW);
                const bool inb = yin && xin;
                const int sb = (r * 8) * kW + xsel;     // plane picked via ds imm offset
                const float dsv = lds[sb];
                const float n0  = lds[sb + 1 * kW];
                const float n1  = lds[sb + 2 * kW];
                const float n2  = lds[sb + 3 * kW];
                const float cc  = lds[sb + 4 * kW];
                const float g0  = lds[sb + 5 * kW];
                const float g1  = lds[sb + 6 * kW];
                const float g2  = lds[sb + 7 * kW];

                const float den = fmaf(n0, rx[j], fmaf(n1, ry[j], fmaf(n2, rz[j], kEps)));
                const float nom = dsv * fmaf(n0, rx[j] + ox,
                                        fmaf(n1, ry[j] + oy, n2 * (rz[j] + oz)));
                float v = nom * __builtin_amdgcn_rcpf(den);
                v = fminf(fmaxf(v, 0.01f), 10.0f);

                // Reference boundary rule: depth[y,x]; 'ld' d2 (dir==7) uses
                // depth[y, x-o] when only ys is out of range (xs=x-o >= 0).
                float fb = dc[j];
                if (dir == 7) {
                    const float dshift = lds[(1 * 8 + 0) * kW + xsel];  // depth[y, x-o]
                    fb = (xs >= 0) ? dshift : dc[j];
                }
                dv[j]  = inb ? v   : fb;
                cv[j]  = inb ? cc  : 0.0f;
                r0v[j] = inb ? g0  : 0.0f;
                r1v[j] = inb ? g1  : 0.0f;
                r2v[j] = inb ? g2  : 0.0f;
            }
        }

        // Output layout: [b, 24, 5, H, W], channels = (depth, conf, r, g, b).
        const int kc = oi * 8 + dir;
        float* p = out + ((size_t)(b * 24 + kc) * 5) * kHW + (size_t)y * kW + x0;
        __builtin_nontemporal_store(dv,  (v4f*)(p));
        __builtin_nontemporal_store(cv,  (v4f*)(p + (size_t)kHW));
        __builtin_nontemporal_store(r0v, (v4f*)(p + (size_t)2 * kHW));
        __builtin_nontemporal_store(r1v, (v4f*)(p + (size_t)3 * kHW));
        __builtin_nontemporal_store(r2v, (v4f*)(p + (size_t)4 * kHW));
    }
}

extern "C" void kernel_launch(void* const* d_in, const int* in_sizes, int n_in,
                              void* d_out, int out_size, void* d_ws, size_t ws_size,
                              hipStream_t stream)
{
    const float* depth  = (const float*)d_in[0];
    const float* normal = (const float*)d_in[1];
    const float* rgb    = (const float*)d_in[2];
    const float* conf   = (const float*)d_in[3];
    const float* kinv   = (const float*)d_in[4];
    float* out = (float*)d_out;

    const int B = in_sizes[0] / kHW;    // depth is [B,1,H,W]
    dim3 grid(kH, (unsigned)B, 3);      // (row, batch, offset-index)
    dim3 block(kW / 4);                 // 160 threads = 5 wave32s, one full row
    prop_kernel<<<grid, block, 0, stream>>>(depth, normal, rgb, conf, kinv, out);
}